// HaloCosineEmbeddingLoss_74448963109284
// MI455X (gfx1250) — compile-verified
//
#include <hip/hip_runtime.h>
#include <hip/hip_bf16.h>

// HaloCosineEmbeddingLoss for MI455X (gfx1250, wave32).
//
// Pipeline:
//   0) memsetAsync: zero accumulators in d_ws
//   1) mask_kernel: 15x15-disk dilation of (class_labels==class_idx) via 30-bit
//      row bitmasks in LDS -> per-pixel byte mask {bit0=pos, bit1=neg} + counts
//   2) sum_kernel:  S[b,d,{pos,neg}] = sum_p E[b,d,p]*mask via
//      v_wmma_f32_16x16x4_f32 (A = 16 ch x 4 px, B cols {pos,neg})
//   3) final_kernel: per-batch cosine similarity, sum -> d_out[0]
//
// Roofline: dominant traffic = embeddings 8*32*512*512*4 B = 268 MB read once
// -> ~12 us at 23.3 TB/s. Everything else is noise; the sum_kernel inner loop
// is kept branchless/minimal so VMEM stays saturated.

#define HDIM 512
#define WDIM 512
#define HW (HDIM * WDIM)
#define NB 8
#define ND 32
#define PAD 7

#define NCHUNK 128                    // sum_kernel grid.x
#define WPB 8                         // waves per block in sum_kernel
#define SLICE (HW / (NCHUNK * WPB))   // 256 pixels per wave

typedef __attribute__((ext_vector_type(2))) float v2f;
typedef __attribute__((ext_vector_type(8))) float v8f;

// ---------------------------------------------------------------------------
// Kernel 1: labels -> pos/neg byte masks + per-batch counts.
// Structure element se(r,c) = ((r-7.5)^2 + (c-7.5)^2) < 8.5^2 = 72.25,
// reduced to one 15-bit mask per window row (derived exactly from the fp math):
//   r:      0      1      2      3     4..11    12     13     14
//   bits: c4-11  c3-12  c2-13  c1-14  c0-14   c1-14  c2-13  c3-12
// ---------------------------------------------------------------------------
__global__ void mask_kernel(const int* __restrict__ labels,
                            const int* __restrict__ class_idx,
                            unsigned char* __restrict__ masks,
                            float* __restrict__ cnt_pos,
                            float* __restrict__ cnt_neg) {
    __shared__ unsigned int rowbits[30];   // 30 rows x 30 valid bits
    __shared__ int sPos, sNeg;

    const int b     = blockIdx.z;
    const int tileY = blockIdx.y * 16;
    const int tileX = blockIdx.x * 16;
    const int tx    = threadIdx.x;
    const int ty    = threadIdx.y;
    const int tid   = ty * 16 + tx;
    const int cls   = class_idx[0];

    if (tid < 30) rowbits[tid] = 0u;
    if (tid == 0) { sPos = 0; sNeg = 0; }
    __syncthreads();

    const int* lb = labels + (size_t)b * HW;
    for (int k = tid; k < 30 * 30; k += 256) {
        const int r  = k / 30;
        const int c  = k % 30;
        const int gy = tileY + r - PAD;
        const int gx = tileX + c - PAD;
        int f = 0;
        if (gy >= 0 && gy < HDIM && gx >= 0 && gx < WDIM)
            f = (lb[gy * WDIM + gx] == cls) ? 1 : 0;
        if (f) atomicOr(&rowbits[r], 1u << c);
    }
    __syncthreads();

    const unsigned int SE[15] = {
        0x0FF0u, 0x1FF8u, 0x3FFCu, 0x7FFEu,
        0x7FFFu, 0x7FFFu, 0x7FFFu, 0x7FFFu,
        0x7FFFu, 0x7FFFu, 0x7FFFu, 0x7FFFu,
        0x7FFEu, 0x3FFCu, 0x1FF8u
    };

    unsigned int dil = 0;
#pragma unroll
    for (int r = 0; r < 15; ++r)
        dil |= (rowbits[ty + r] >> tx) & SE[r];

    const int lab = (rowbits[ty + PAD] >> (tx + PAD)) & 1;
    const int pos = lab;
    const int neg = (!lab) && (dil != 0u);

    masks[(size_t)b * HW + (size_t)(tileY + ty) * WDIM + (tileX + tx)] =
        (unsigned char)(pos | (neg << 1));

    atomicAdd(&sPos, pos);
    atomicAdd(&sNeg, neg);
    __syncthreads();
    if (tid == 0) {
        atomicAdd(&cnt_pos[b], (float)sPos);
        atomicAdd(&cnt_neg[b], (float)sNeg);
    }
}

// ---------------------------------------------------------------------------
// Kernel 2: masked channel sums via V_WMMA_F32_16X16X4_F32.
// A (16x4 f32): lane m (<16) holds E[ch=cb*16+m][p..p+1]; lane m+16 holds
//               E[ch][p+2..p+3]  (ISA A-layout: VGPR0=K0/K2, VGPR1=K1/K3).
// B (4x16 f32): column 0 = pos mask, column 1 = neg mask, rest 0.
//               Built branchlessly: lane's bit position sh0 = 8*koff + (col&1)
//               and validity mask (col<2) -> lshr/and/cvt only, EXEC untouched.
// C (16x16 f32): col0 accumulates sum(E*pos), col1 sum(E*neg).
// C layout: lane n (<16) VGPR j holds M=j,N=n; lane n+16 holds M=j+8,N=n.
// ---------------------------------------------------------------------------
__global__ void sum_kernel(const float* __restrict__ emb,
                           const unsigned char* __restrict__ masks,
                           float* __restrict__ S_pos,
                           float* __restrict__ S_neg) {
    const int b     = blockIdx.z;
    const int cb    = blockIdx.y;          // channel block: 0 -> ch 0..15, 1 -> 16..31
    const int wave  = threadIdx.x >> 5;
    const int lane  = threadIdx.x & 31;
    const int col   = lane & 15;           // A row (channel) and B/C column
    const int khalf = lane >> 4;           // 0: K=0,1   1: K=2,3
    const int koff  = khalf * 2;

    const int p0 = (blockIdx.x * WPB + wave) * SLICE;

    const float* base = emb + ((size_t)b * ND + (size_t)(cb * 16 + col)) * HW;
    const unsigned char* mbase = masks + (size_t)b * HW;

    // Branchless B-operand selectors (no EXEC manipulation in the hot loop).
    const unsigned int valid = (col < 2) ? 1u : 0u;
    const int sh0 = 8 * koff + (col & 1);  // pos -> bit0, neg -> bit1 (<= 17)
    const int sh1 = sh0 + 8;               // next mask byte            (<= 25)

    v8f acc0 = {};
    v8f acc1 = {};
    for (int i = 0; i < SLICE; i += 16) {
        const int p = p0 + i;
        union { uint4 v; unsigned int w[4]; } mu;
        mu.v = *(const uint4*)(mbase + p);             // 16 mask bytes, broadcast
        __builtin_prefetch(base + p + 256, 0, 0);      // global_prefetch_b8, 1KB ahead
#pragma unroll
        for (int u = 0; u < 4; ++u) {
            const v2f a = *(const v2f*)(base + p + 4 * u + koff);   // 8B per lane
            const unsigned int mw = mu.w[u];
            const float b0 = (float)((mw >> sh0) & valid);
            const float b1 = (float)((mw >> sh1) & valid);
            const v2f bb = {b0, b1};
            if (u & 1)
                acc1 = __builtin_amdgcn_wmma_f32_16x16x4_f32(
                    false, a, false, bb, (short)0, acc1, false, false);
            else
                acc0 = __builtin_amdgcn_wmma_f32_16x16x4_f32(
                    false, a, false, bb, (short)0, acc0, false, false);
        }
    }
    const v8f acc = acc0 + acc1;

    if (col < 2) {                       // lanes 0,16 -> pos ; lanes 1,17 -> neg
        float* S = (col == 0) ? S_pos : S_neg;
        const int chbase = cb * 16 + khalf * 8;
#pragma unroll
        for (int j = 0; j < 8; ++j)
            atomicAdd(&S[b * ND + chbase + j], acc[j]);
    }
}

// ---------------------------------------------------------------------------
// Kernel 3: per-batch cosine similarity, then sum over batches.
// ---------------------------------------------------------------------------
__global__ void final_kernel(const float* __restrict__ S_pos,
                             const float* __restrict__ S_neg,
                             const float* __restrict__ cnt_pos,
                             const float* __restrict__ cnt_neg,
                             float* __restrict__ out) {
    __shared__ float cossim[NB];
    const int t = threadIdx.x;
    if (t < NB) {
        const float ip = 1.0f / cnt_pos[t];
        const float in = 1.0f / cnt_neg[t];
        float dot = 0.0f, np = 0.0f, nn = 0.0f;
        for (int d = 0; d < ND; ++d) {
            const float rp = S_pos[t * ND + d] * ip;
            const float rn = S_neg[t * ND + d] * in;
            dot += rp * rn;
            np  += rp * rp;
            nn  += rn * rn;
        }
        const float denom = fmaxf(sqrtf(np) * sqrtf(nn), 1e-8f);
        cossim[t] = dot / denom;
    }
    __syncthreads();
    if (t == 0) {
        float s = 0.0f;
        for (int i = 0; i < NB; ++i) s += cossim[i];
        out[0] = s;
    }
}

// ---------------------------------------------------------------------------
// Workspace layout:
//   [0,    1024)  S_pos   (8*32 f32)
//   [1024, 2048)  S_neg   (8*32 f32)
//   [2048, 2080)  cnt_pos (8 f32)
//   [2080, 2112)  cnt_neg (8 f32)
//   [4096, 4096 + 8*512*512) mask bytes
// ---------------------------------------------------------------------------
extern "C" void kernel_launch(void* const* d_in, const int* in_sizes, int n_in,
                              void* d_out, int out_size, void* d_ws, size_t ws_size,
                              hipStream_t stream) {
    (void)in_sizes; (void)n_in; (void)out_size; (void)ws_size;

    const float* emb       = (const float*)d_in[0];
    const int*   labels    = (const int*)d_in[1];
    const int*   class_idx = (const int*)d_in[2];
    float*       out       = (float*)d_out;

    char* ws = (char*)d_ws;
    float* S_pos   = (float*)(ws);
    float* S_neg   = (float*)(ws + 1024);
    float* cnt_pos = (float*)(ws + 2048);
    float* cnt_neg = (float*)(ws + 2080);
    unsigned char* masks = (unsigned char*)(ws + 4096);

    hipMemsetAsync(d_ws, 0, 4096, stream);

    mask_kernel<<<dim3(WDIM / 16, HDIM / 16, NB), dim3(16, 16), 0, stream>>>(
        labels, class_idx, masks, cnt_pos, cnt_neg);

    sum_kernel<<<dim3(NCHUNK, 2, NB), WPB * 32, 0, stream>>>(
        emb, masks, S_pos, S_neg);

    final_kernel<<<1, 32, 0, stream>>>(S_pos, S_neg, cnt_pos, cnt_neg, out);
}